// ContrastiveLoss_21775484191011
// MI455X (gfx1250) — compile-verified
//
#include <hip/hip_runtime.h>
#include <hip/hip_bf16.h>
#include <math.h>

// ContrastiveLoss on MI455X (gfx1250, wave32).
// Single pass over x (134 MB) computing per-row {dot(row, anchor), ||row||^2}
// simultaneously via V_WMMA_F32_16X16X4_F32 Gram accumulation:
//   - per wave: tile rows = [anchor, 15 data rows], K-slice of 512 cols
//   - A fragment (16x4 f32) == B fragment (4x16 f32, = A^T) element-for-element
//     per the ISA VGPR layouts, so c = wmma(a, a, c) accumulates D = A * A^T.
//   - D[:,0] = dots vs anchor, diag(D) = squared norms.
// Then a tiny deterministic reduction kernel produces the scalar loss.

typedef __attribute__((ext_vector_type(2))) float v2f;
typedef __attribute__((ext_vector_type(8))) float v8f;

#define N_ROWS 16384
#define D_COLS 2048
#define KSPLIT 4
#define KSLICE (D_COLS / KSPLIT) /* 512 */
#define CHUNKS (KSLICE / 4)      /* 128 wmma steps per wave */
#define RPT 15                   /* data rows per tile (row 0 = anchor) */
#define TILES (((N_ROWS - 1) + RPT - 1) / RPT) /* 1093 */
#define WPB 8                    /* waves per block */

__global__ __launch_bounds__(WPB * 32)
void gram_kernel(const float* __restrict__ x,
                 const long long* __restrict__ pos_pair,
                 float* __restrict__ P,     /* [TILES*RPT][KSPLIT][2] dot,sq */
                 float* __restrict__ sq0p)  /* [KSPLIT] anchor sq partials */
{
    __shared__ float scr[WPB][16 * 17]; /* padded 16x16 D dump, no bank conflicts */

    const int lane = threadIdx.x & 31;
    const int wv   = threadIdx.x >> 5;
    const int gw   = blockIdx.x * WPB + wv;
    const int t    = gw / KSPLIT;   /* row tile */
    const int ks   = gw % KSPLIT;   /* K slice  */
    const int anchor = (int)pos_pair[0];

    /* Tile row 0 -> anchor; tile rows 1..15 enumerate rows != anchor. */
    const int slot = lane & 15;
    int row;
    if (slot == 0) {
        row = anchor;
    } else {
        int idx = t * RPT + (slot - 1);
        row = (idx < N_ROWS - 1) ? (idx + (idx >= anchor ? 1 : 0)) : anchor;
    }

    /* A-fragment addressing: lanes 0-15 hold K=0,1; lanes 16-31 hold K=2,3. */
    const float* p = x + (size_t)row * D_COLS + (size_t)ks * KSLICE + ((lane >> 4) << 1);

    v8f c = {};
#pragma unroll 8
    for (int i = 0; i < CHUNKS; ++i) {
        v2f a = *(const v2f*)p;
        if ((i & 7) == 0) __builtin_prefetch(p + 128, 0, 1); /* global_prefetch_b8 */
        /* D += A * A^T : B fragment is bit-identical to A fragment. */
        c = __builtin_amdgcn_wmma_f32_16x16x4_f32(
                /*neg_a=*/false, a, /*neg_b=*/false, a,
                /*c_mod=*/(short)0, c, /*reuse_a=*/false, /*reuse_b=*/false);
        p += 4;
    }

    /* Dump D (16x16) to LDS: lanes 0-15 hold M=v, lanes 16-31 hold M=v+8. */
    float* s = scr[wv];
    const int M0 = (lane < 16) ? 0 : 8;
    const int Nc = lane & 15;
#pragma unroll
    for (int v = 0; v < 8; ++v) s[(M0 + v) * 17 + Nc] = c[v];
    __syncthreads();

    if (t < TILES) {
        if (lane >= 1 && lane <= 15) {
            float dot = s[lane * 17 + 0];    /* D[lane][0] = dot(row, anchor) */
            float sq  = s[lane * 17 + lane]; /* D[lane][lane] = ||row||^2     */
            size_t o = (((size_t)t * RPT + (lane - 1)) * KSPLIT + ks) * 2;
            P[o]     = dot;
            P[o + 1] = sq;
        }
        if (t == 0 && lane == 0) sq0p[ks] = s[0]; /* D[0][0] = ||anchor||^2 */
    }
}

__global__ __launch_bounds__(1024)
void finish_kernel(const float* __restrict__ P,
                   const float* __restrict__ sq0p,
                   const long long* __restrict__ pos_pair,
                   float* __restrict__ out)
{
    __shared__ float red[1024];
    __shared__ float cosj_s;

    const int tid    = threadIdx.x;
    const int anchor = (int)pos_pair[0];
    const int jrow   = (int)pos_pair[1];

    float sq0 = 0.f;
#pragma unroll
    for (int kk = 0; kk < KSPLIT; ++kk) sq0 += sq0p[kk];
    const float ni = fmaxf(sqrtf(sq0), 1e-8f);

    float acc = 0.f;
    for (int idx = tid; idx < N_ROWS - 1; idx += 1024) {
        size_t o = (size_t)idx * KSPLIT * 2;
        float dot = 0.f, sq = 0.f;
#pragma unroll
        for (int kk = 0; kk < KSPLIT; ++kk) {
            dot += P[o + 2 * kk];
            sq  += P[o + 2 * kk + 1];
        }
        float nk  = fmaxf(sqrtf(sq), 1e-8f);
        float cos = dot / (nk * ni);
        acc += expf(cos * 10.0f); /* cos / TEMP, TEMP = 0.1 */
        int row = idx + (idx >= anchor ? 1 : 0);
        if (row == jrow) cosj_s = cos;
    }

    red[tid] = acc;
    __syncthreads();
    for (int off = 512; off > 0; off >>= 1) {
        if (tid < off) red[tid] += red[tid + off];
        __syncthreads();
    }
    if (tid == 0) {
        /* loss = -log(e_j / (den + 1e-6)) = log(den + 1e-6) - cos_j / TEMP */
        out[0] = logf(red[0] + 1e-6f) - 10.0f * cosj_s;
    }
}

extern "C" void kernel_launch(void* const* d_in, const int* in_sizes, int n_in,
                              void* d_out, int out_size, void* d_ws, size_t ws_size,
                              hipStream_t stream) {
    const float*     x  = (const float*)d_in[0];
    const long long* pp = (const long long*)d_in[1]; /* int64 pos_pair (i, j) */

    float* P    = (float*)d_ws;                               /* ~524 KB */
    float* sq0p = P + (size_t)TILES * RPT * KSPLIT * 2;       /* +4 floats */

    const int total_waves = TILES * KSPLIT;                   /* 4372 */
    const int blocks = (total_waves + WPB - 1) / WPB;         /* 547  */

    gram_kernel<<<blocks, WPB * 32, 0, stream>>>(x, pp, P, sq0p);
    finish_kernel<<<1, 1024, 0, stream>>>(P, sq0p, pp, (float*)d_out);
}